// MyLoss_1580547974259
// MI455X (gfx1250) — compile-verified
//
#include <hip/hip_runtime.h>
#include <hip/hip_bf16.h>

// ---------------------------------------------------------------------------
// Problem constants (from reference)
// ---------------------------------------------------------------------------
#define N_ROWS   100000
#define N_BINS   512
#define BATCH    8192
#define KNN      16
#define ROWCHUNKS 32            // row chunks for the column-sum partials

typedef __attribute__((ext_vector_type(2))) float v2f;
typedef __attribute__((ext_vector_type(8))) float v8f;

// ---------------------------------------------------------------------------
// Kernel 1: column sums of outputs[0:8192, :] via WMMA f32 16x16x4.
//   b_col = ones(1, 8192) @ outputs[0:8192]  -- a real matmul with a ones A.
// Grid: 1024 blocks x 32 threads (one wave). block = (rowchunk, colgroup).
// Each wave accumulates 256 rows x 16 cols with 64 v_wmma_f32_16x16x4_f32.
// A = all ones  =>  every row of C equals the 4-row column sum; exact f32.
// ---------------------------------------------------------------------------
__global__ void colsum_wmma_kernel(const float* __restrict__ outputs,
                                   float* __restrict__ colpart /* [32][512] */) {
    const int rc   = blockIdx.x >> 5;   // 0..31  row chunk (256 rows)
    const int cg   = blockIdx.x & 31;   // 0..31  column group (16 bins)
    const int col0 = cg << 4;
    const int lane = threadIdx.x;       // 0..31, full wave (EXEC all ones)
    const int n    = lane & 15;         // column within group (B/C striping)
    const int rh   = (lane >> 4) << 1;  // 0 or 2: which pair of rows this half-wave feeds

    const v2f a = {1.0f, 1.0f};         // A(16x4) all-ones: 2 VGPRs per lane
    v8f c = {};                         // C/D 16x16 f32 accumulator (8 VGPRs)

    const int row0 = rc << 8;           // rc * 256
    const float* p = outputs + (size_t)(row0 + rh) * N_BINS + col0 + n;

    #pragma unroll 4
    for (int it = 0; it < 64; ++it) {   // 64 chunks of 4 rows
        v2f bv;
        bv.x = p[0];                    // row r0+rh,   col col0+n
        bv.y = p[N_BINS];               // row r0+rh+1, col col0+n
        c = __builtin_amdgcn_wmma_f32_16x16x4_f32(
                /*neg_a=*/false, a, /*neg_b=*/false, bv,
                /*c_mod=*/(short)0, c, /*reuse_a=*/false, /*reuse_b=*/false);
        p += 4 * N_BINS;
    }

    // All 16 rows of C are identical column sums; row M=0 lives in c[0],
    // lanes 0..15 carry N=0..15 (lanes 16..31 duplicate via M=8).
    if (lane < 16)
        colpart[rc * N_BINS + col0 + n] = c[0];
}

// ---------------------------------------------------------------------------
// Kernel 2: gather + max-plus. One block (256 threads = 8 waves) per batch
// element. Base row staged into LDS with CDNA5 async global->LDS loads
// (ASYNCcnt). Each wave handles 2 neighbors; neighbor rows streamed with
// coalesced b128 loads; per-wave max reduction via 32-lane shuffles.
// Writes booster_weights (d_out[3+b]) and weighted add-sum partials (ws).
// ---------------------------------------------------------------------------
__global__ void gather_maxplus_kernel(const float* __restrict__ outputs,
                                      const float* __restrict__ y,
                                      const float* __restrict__ wgt,
                                      float* __restrict__ booster,   // d_out + 3
                                      float* __restrict__ spart) {   // [8192]
    __shared__ float base[N_BINS];
    __shared__ float addv[KNN];

    const int b   = blockIdx.x;
    const int tid = threadIdx.x;

    // ---- async copy base row outputs[b, :] (2 KB) into LDS -----------------
    {
        unsigned lds_off = (unsigned)(size_t)(&base[tid * 2]);       // LDS byte addr
        const float* gsrc = outputs + (size_t)b * N_BINS + tid * 2;  // 8B per lane
        asm volatile("global_load_async_to_lds_b64 %0, %1, off"
                     :: "v"(lds_off), "v"(gsrc) : "memory");
    }
    asm volatile("s_wait_asynccnt 0" ::: "memory");
    __syncthreads();

    const int wave = tid >> 5;
    const int lane = tid & 31;

    for (int kk = 0; kk < 2; ++kk) {
        const int k  = (wave << 1) | kk;                 // 0..15
        const int nn = (int)y[(size_t)b * KNN + k];      // trunc (y >= 0)
        const float* row = outputs + (size_t)nn * N_BINS;

        float m = -1.0e30f;
        #pragma unroll
        for (int i = 0; i < 4; ++i) {
            const int cidx = lane * 4 + i * 128;         // coalesced 512B/wave
            float4 v = *reinterpret_cast<const float4*>(row + cidx);
            float m0 = fmaxf(v.x + base[cidx + 0], v.y + base[cidx + 1]);
            float m1 = fmaxf(v.z + base[cidx + 2], v.w + base[cidx + 3]);
            m = fmaxf(m, fmaxf(m0, m1));
        }
        // wave32 max reduction
        #pragma unroll
        for (int off = 16; off >= 1; off >>= 1)
            m = fmaxf(m, __shfl_xor(m, off, 32));
        if (lane == 0) addv[k] = m;
    }
    __syncthreads();

    if (tid == 0) {
        float s = 0.0f;
        #pragma unroll
        for (int k = 0; k < KNN; ++k) s += addv[k];
        booster[b] = fmaxf((2.0f - s * (1.0f / KNN)) * 0.5f, 0.5f);
        spart[b]   = s * wgt[b];   // later: /(BATCH*KNN) for add_mean
    }
}

// ---------------------------------------------------------------------------
// Kernel 3: single-block deterministic finalize.
//   b_col from column partials -> max/min -> b; sum weighted add partials ->
//   add_mean -> diff; write cost, diff, ratio.
// ---------------------------------------------------------------------------
__global__ void finalize_kernel(const float* __restrict__ colpart,
                                const float* __restrict__ spart,
                                float* __restrict__ out) {
    __shared__ float red[N_BINS];
    __shared__ float s_bmax, s_bmin;
    const int t = threadIdx.x;          // 512 threads

    float cs = 0.0f;
    #pragma unroll
    for (int rc = 0; rc < ROWCHUNKS; ++rc) cs += colpart[rc * N_BINS + t];

    // max reduce
    red[t] = cs; __syncthreads();
    for (int s = 256; s > 0; s >>= 1) {
        if (t < s) red[t] = fmaxf(red[t], red[t + s]);
        __syncthreads();
    }
    if (t == 0) s_bmax = red[0];
    __syncthreads();

    // min reduce
    red[t] = cs; __syncthreads();
    for (int s = 256; s > 0; s >>= 1) {
        if (t < s) red[t] = fminf(red[t], red[t + s]);
        __syncthreads();
    }
    if (t == 0) s_bmin = red[0];
    __syncthreads();

    // sum of weighted add partials (fixed order: strided 16 per thread + tree)
    float ps = 0.0f;
    #pragma unroll
    for (int i = 0; i < BATCH / N_BINS; ++i) ps += spart[t + i * N_BINS];
    red[t] = ps; __syncthreads();
    for (int s = 256; s > 0; s >>= 1) {
        if (t < s) red[t] += red[t + s];
        __syncthreads();
    }

    if (t == 0) {
        const float add_mean = red[0] / (float)(BATCH * KNN);
        const float d        = (2.0f - add_mean);
        const float diff     = d * d;
        const float ratio    = (s_bmax - s_bmin) / ((float)N_ROWS / (float)N_BINS);
        out[0] = ratio + diff;   // cost
        out[1] = diff;
        out[2] = ratio;
    }
}

// ---------------------------------------------------------------------------
// Host-side launcher
// ---------------------------------------------------------------------------
extern "C" void kernel_launch(void* const* d_in, const int* in_sizes, int n_in,
                              void* d_out, int out_size, void* d_ws, size_t ws_size,
                              hipStream_t stream) {
    const float* outputs = (const float*)d_in[0];   // (100000, 512) f32
    const float* y       = (const float*)d_in[1];   // (8192, 16)   f32 indices
    const float* wgt     = (const float*)d_in[2];   // (8192,)      f32
    float* out = (float*)d_out;                     // [cost, diff, ratio, booster(8192)]

    float* colpart = (float*)d_ws;                  // 32*512 floats (64 KB)
    float* spart   = colpart + ROWCHUNKS * N_BINS;  // 8192 floats  (32 KB)

    colsum_wmma_kernel  <<<ROWCHUNKS * 32, 32,  0, stream>>>(outputs, colpart);
    gather_maxplus_kernel<<<BATCH,        256, 0, stream>>>(outputs, y, wgt,
                                                            out + 3, spart);
    finalize_kernel     <<<1,             512, 0, stream>>>(colpart, spart, out);
}